// SoftmaxAttentionModel_33002528703339
// MI455X (gfx1250) — compile-verified
//
#include <hip/hip_runtime.h>
#include <math.h>

typedef __attribute__((ext_vector_type(2))) float v2f;
typedef __attribute__((ext_vector_type(8))) float v8f;
typedef unsigned int u32x4 __attribute__((ext_vector_type(4)));
typedef int i32x4 __attribute__((ext_vector_type(4)));
typedef int i32x8 __attribute__((ext_vector_type(8)));

#define BS 64
#define SL 8192
#define E 64
#define DD 64
#define N_VALUES 64
#define VOCAB 1088          // N_KEYS + N_VALUES
#define INK 128             // E + N_VALUES
#define NCHUNK 4
#define CHUNK (SL / NCHUNK) // 2048
#define T2 256
#define PER_T (CHUNK / T2)  // 8
#define PART_STRIDE 66      // [m, Z, y[64]]

#ifdef __has_builtin
#if __has_builtin(__builtin_amdgcn_tensor_load_to_lds) && \
    __has_builtin(__builtin_amdgcn_s_wait_tensorcnt)
#define USE_TDM 1
#endif
#endif
#ifndef USE_TDM
#define USE_TDM 0
#endif

// ---------------------------------------------------------------------------
// Kernel 0: per-batch projection.
//   Q[b,d] = embK[q[b]] . Wq_w[d,:] + Wq_b[d]
//   u[b,i] = sum_d Wk_w[d,i] * Q[b,d]      (i in [0,128))
// u[0:64]   -> one-hot(value) score contribution (direct lookup)
// u[64:128] -> key-embedding projection vector (feeds the WMMA GEMM)
// Wk_b . Q is uniform over l within a batch -> cancels in softmax, dropped.
// ---------------------------------------------------------------------------
__global__ void k_project(const int* __restrict__ q,
                          const float* __restrict__ embK,
                          const float* __restrict__ Wk_w,
                          const float* __restrict__ Wq_w,
                          const float* __restrict__ Wq_b,
                          float* __restrict__ ws_u) {
  int b = blockIdx.x;
  int tid = threadIdx.x; // 128 threads
  __shared__ float Qs[DD];
  int qi = q[b];
  if (tid < DD) {
    float acc = Wq_b[tid];
    const float* er = embK + (size_t)qi * E;
    const float* wr = Wq_w + (size_t)tid * E;
    #pragma unroll 8
    for (int e = 0; e < E; ++e) acc = fmaf(er[e], wr[e], acc);
    Qs[tid] = acc;
  }
  __syncthreads();
  float acc = 0.f;
  #pragma unroll 8
  for (int d = 0; d < DD; ++d) acc = fmaf(Wk_w[d * INK + tid], Qs[d], acc);
  ws_u[b * INK + tid] = acc;
}

// ---------------------------------------------------------------------------
// Kernel 1: key score table via fp32 WMMA.
//   T[b,k] = sum_e embK[k,e] * u[b,64+e]
// GEMM: M=VOCAB(1088)=68 tiles, N=BS(64)=4 tiles, K=64 in 16 steps of 4.
// One wave per 16x16 output tile; 272 waves total.
// A 16x4 f32 layout (ISA 7.12.2): lanes 0-15 -> K={0,1}, lanes 16-31 -> K={2,3},
// M = lane%16. B 4x16: row-striped across lanes, same half-wave K split.
// C/D: VGPR r holds rows r (lanes 0-15) and r+8 (lanes 16-31).
// ---------------------------------------------------------------------------
__global__ void k_table_wmma(const float* __restrict__ embK,
                             const float* __restrict__ ws_u,
                             float* __restrict__ ws_T) {
  int lane = threadIdx.x & 31;
  int wave = threadIdx.x >> 5;
  int gw = blockIdx.x * 8 + wave;   // 0..271
  int mt = gw % 68;                 // key-row tile
  int nt = gw / 68;                 // batch-col tile
  int mrow = mt * 16 + (lane & 15);
  int ncol = nt * 16 + (lane & 15);
  int koff = (lane < 16) ? 0 : 2;

  v8f c = {};
  #pragma unroll
  for (int kk = 0; kk < E; kk += 4) {
    v2f a, bv;
    a.x = embK[mrow * E + kk + koff];
    a.y = embK[mrow * E + kk + koff + 1];
    bv.x = ws_u[ncol * INK + 64 + kk + koff];
    bv.y = ws_u[ncol * INK + 64 + kk + koff + 1];
    c = __builtin_amdgcn_wmma_f32_16x16x4_f32(
        /*neg_a=*/false, a, /*neg_b=*/false, bv,
        /*c_mod=*/(short)0, c, /*reuse_a=*/false, /*reuse_b=*/false);
  }

  int rbase = mt * 16 + ((lane >> 4) << 3);
  int col = nt * 16 + (lane & 15);
  #pragma unroll
  for (int r = 0; r < 8; ++r)
    ws_T[(size_t)col * VOCAB + rbase + r] = c[r];
}

// ---------------------------------------------------------------------------
// Kernel 2: streaming softmax pass. One block per (chunk, batch).
// Reads x exactly once (coalesced). Scores cached in LDS for the 2nd pass.
// t-table staged into LDS via the Tensor Data Mover (TENSORcnt path), with a
// redundant same-value vector staging loop so correctness does not depend on
// TDM descriptor semantics. Emits per-chunk partial (max, Z, y[64]).
// ---------------------------------------------------------------------------
__global__ void k_attend(const int* __restrict__ x,
                         const float* __restrict__ ws_u,
                         const float* __restrict__ ws_T,
                         float* __restrict__ ws_part) {
  int chunk = blockIdx.x;
  int b = blockIdx.y;
  int tid = threadIdx.x;

  // Declaration order: arrays that are fully rewritten after the first
  // barrier (sc_s, val_s) sit right after t_s, so a mis-sized TDM tile
  // cannot corrupt live state below.
  __shared__ float t_s[VOCAB];
  __shared__ float sc_s[CHUNK];
  __shared__ unsigned char val_s[CHUNK];
  __shared__ float uv_s[N_VALUES];
  __shared__ float red[8];
  __shared__ float ybuf[N_VALUES];

#if USE_TDM
  if (tid < 32) {
    // D# group0: count=1, lds_addr, global_addr(57b), type=2 ("image")
    unsigned lds_addr = (unsigned)(size_t)(void*)t_s;
    unsigned long long ga =
        (unsigned long long)(size_t)(const void*)(ws_T + (size_t)b * VOCAB);
    u32x4 g0 = {0u, 0u, 0u, 0u};
    g0[0] = 1u;                                   // count=1, no gather
    g0[1] = lds_addr;                             // [63:32] lds_addr
    g0[2] = (unsigned)ga;                         // [95:64] global_addr lo
    g0[3] = (unsigned)((ga >> 32) & 0x01ffffffull)// [120:96] global_addr hi
            | 0x80000000u;                        // [127:126] type=2
    // D# group1: data_size=4B; tensor 1088x1, tile 1088x1, stride 1088
    i32x8 g1 = {0, 0, 0, 0, 0, 0, 0, 0};
    g1[0] = 0x00020000;                           // data_size=2 (4 bytes)
    g1[1] = (int)(VOCAB << 16);                   // tensor_dim0[15:0]
    g1[2] = (int)(1 << 16);                       // tensor_dim0 hi=0, dim1=1
    g1[3] = (int)(VOCAB << 16);                   // tile_dim0=1088
    g1[4] = 1;                                    // tile_dim1=1, tile_dim2=0
    g1[5] = VOCAB;                                // tensor_dim0_stride
    // D# groups 2/3: higher dims = length 1, unused tiles = 0
    i32x4 g2 = {1, 1, 0, 0};                      // tensor_dim2=1, dim3=1
    i32x4 g3 = {0, 0x00010000, 0, 0};             // tensor_dim4=1
    // 6-arg toolchain variant: extra (unused / VADDR4-slot) group, zeros.
    i32x8 g4 = {0, 0, 0, 0, 0, 0, 0, 0};
    __builtin_amdgcn_tensor_load_to_lds(g0, g1, g2, g3, g4, 0);
    __builtin_amdgcn_s_wait_tensorcnt(0);
  }
#endif
  // Always-correct staging (same values; overwrite is benign).
  for (int i = tid; i < VOCAB; i += T2) t_s[i] = ws_T[(size_t)b * VOCAB + i];
  if (tid < N_VALUES) { uv_s[tid] = ws_u[b * INK + tid]; ybuf[tid] = 0.f; }
  __syncthreads();

  const int* kp = x + (size_t)b * 2 * SL + chunk * CHUNK;
  const int* vp = kp + SL;

  // pass 1: scores + running max
  float m = -INFINITY;
  #pragma unroll
  for (int j = 0; j < PER_T; ++j) {
    int i = tid + j * T2;
    int ki = kp[i];
    int vi = vp[i] - N_VALUES;
    float s = (t_s[ki] + uv_s[vi]) * 0.125f;  // 1/sqrt(D)
    sc_s[i] = s;
    val_s[i] = (unsigned char)vi;
    m = fmaxf(m, s);
  }
  #pragma unroll
  for (int off = 16; off > 0; off >>= 1)
    m = fmaxf(m, __shfl_xor(m, off, 32));
  if ((tid & 31) == 0) red[tid >> 5] = m;
  __syncthreads();
  m = red[0];
  #pragma unroll
  for (int w = 1; w < 8; ++w) m = fmaxf(m, red[w]);

  // pass 2: exp + Z + 64-bucket LDS histogram
  float z = 0.f;
  #pragma unroll
  for (int j = 0; j < PER_T; ++j) {
    int i = tid + j * T2;
    float e = __expf(sc_s[i] - m);
    z += e;
    atomicAdd(&ybuf[val_s[i]], e);
  }
  #pragma unroll
  for (int off = 16; off > 0; off >>= 1)
    z += __shfl_xor(z, off, 32);
  __syncthreads();               // red[] reads done, ybuf atomics done
  if ((tid & 31) == 0) red[tid >> 5] = z;
  __syncthreads();
  float Z = 0.f;
  #pragma unroll
  for (int w = 0; w < 8; ++w) Z += red[w];

  float* part = ws_part + ((size_t)b * NCHUNK + chunk) * PART_STRIDE;
  if (tid == 0) { part[0] = m; part[1] = Z; }
  if (tid < N_VALUES) part[2 + tid] = ybuf[tid];
}

// ---------------------------------------------------------------------------
// Kernel 3: merge chunk partials. One block per batch, one thread per bucket.
// ---------------------------------------------------------------------------
__global__ void k_combine(const float* __restrict__ ws_part,
                          float* __restrict__ out) {
  int b = blockIdx.x;
  int n = threadIdx.x; // 64
  const float* p = ws_part + (size_t)b * NCHUNK * PART_STRIDE;
  float M = -INFINITY;
  #pragma unroll
  for (int c = 0; c < NCHUNK; ++c) M = fmaxf(M, p[c * PART_STRIDE]);
  float Z = 0.f, Y = 0.f;
  #pragma unroll
  for (int c = 0; c < NCHUNK; ++c) {
    float w = __expf(p[c * PART_STRIDE] - M);
    Z += p[c * PART_STRIDE + 1] * w;
    Y += p[c * PART_STRIDE + 2 + n] * w;
  }
  out[b * N_VALUES + n] = Y / Z;
}

// ---------------------------------------------------------------------------
extern "C" void kernel_launch(void* const* d_in, const int* in_sizes, int n_in,
                              void* d_out, int out_size, void* d_ws, size_t ws_size,
                              hipStream_t stream) {
  const int*   x    = (const int*)d_in[0];
  const int*   q    = (const int*)d_in[1];
  const float* embK = (const float*)d_in[2];
  const float* Wk_w = (const float*)d_in[3];
  // d_in[4] = Wk_b: uniform additive score shift per batch -> cancels in softmax
  const float* Wq_w = (const float*)d_in[5];
  const float* Wq_b = (const float*)d_in[6];
  float* out = (float*)d_out;

  // workspace layout (floats): u[64][128] | T[64][1088] | part[64][4][66]
  float* ws_u    = (float*)d_ws;                 // 32 KB
  float* ws_T    = ws_u + BS * INK;              // 278 KB
  float* ws_part = ws_T + (size_t)BS * VOCAB;    // 67 KB   (~380 KB total)

  k_project<<<BS, INK, 0, stream>>>(q, embK, Wk_w, Wq_w, Wq_b, ws_u);
  k_table_wmma<<<34, 256, 0, stream>>>(embK, ws_u, ws_T);   // 272 waves
  dim3 g2(NCHUNK, BS);
  k_attend<<<g2, T2, 0, stream>>>(x, ws_u, ws_T, ws_part);
  k_combine<<<BS, N_VALUES, 0, stream>>>(ws_part, out);
}